// ResGatedGCN_36687610642605
// MI455X (gfx1250) — compile-verified
//
#include <hip/hip_runtime.h>
#include <math.h>

typedef __attribute__((ext_vector_type(2))) float f32x2;
typedef __attribute__((ext_vector_type(8))) float f32x8;

#define NNODE 50000
#define NEDGE 640000
#define HDIM  128
#define NCLASS 40
#define NLAYER 3

// ---------------------------------------------------------------------------
// GEMM: out[M, ncols] = A[M,128] @ W[128, ncols] + bias, f32 WMMA 16x16x4.
// 256 threads = 8 waves; each wave owns a 16-row stripe, NT 16-col tiles.
// W transposed into LDS (pad-132 stride -> conflict-free ds_load_b64 frags).
// ---------------------------------------------------------------------------
template<int NT>
__global__ __launch_bounds__(256)
void gemm_k128_wmma(const float* __restrict__ A, const float* __restrict__ W,
                    const float* __restrict__ bias, float* __restrict__ out,
                    int M, int ncols) {
  __shared__ float sA[128 * 132];
  __shared__ float sWt[NT * 16 * 132];
  const int tid  = threadIdx.x;
  const int lane = tid & 31;
  const int wave = tid >> 5;
  const int rowBlk = blockIdx.x * 128;

  // Stage W transposed: sWt[n*132 + k] = W[k*ncols + n]  (coalesced global b128)
  const int n4cnt = ncols >> 2;
  for (int idx4 = tid; idx4 < 128 * n4cnt; idx4 += 256) {
    int k = idx4 / n4cnt;
    int n = (idx4 - k * n4cnt) << 2;
    const float4 w = *(const float4*)&W[(size_t)k * ncols + n];
    sWt[(n + 0) * 132 + k] = w.x;
    sWt[(n + 1) * 132 + k] = w.y;
    sWt[(n + 2) * 132 + k] = w.z;
    sWt[(n + 3) * 132 + k] = w.w;
  }
  // Zero-pad columns ncols..NT*16-1 (MLP case)
  for (int idx = tid; idx < (NT * 16 - ncols) * 128; idx += 256) {
    int n = ncols + (idx >> 7);
    int k = idx & 127;
    sWt[n * 132 + k] = 0.0f;
  }
  // Stage 128-row A tile (zero-fill rows >= M)
  #pragma unroll
  for (int i = 0; i < 16; ++i) {
    int idx4 = tid + i * 256;
    int r = idx4 >> 5;
    int c = (idx4 & 31) << 2;
    int row = rowBlk + r;
    float4 v = make_float4(0.f, 0.f, 0.f, 0.f);
    if (row < M) v = *(const float4*)&A[(size_t)row * 128 + c];
    *(float4*)&sA[r * 132 + c] = v;
  }
  __syncthreads();

  f32x8 acc[NT];
  {
    f32x8 z = {0.f, 0.f, 0.f, 0.f, 0.f, 0.f, 0.f, 0.f};
    #pragma unroll
    for (int t = 0; t < NT; ++t) acc[t] = z;
  }

  const int l16   = lane & 15;
  const int khalf = (lane >> 4) << 1;   // lanes 0-15: K0/K1, lanes 16-31: K2/K3
  const int aRow  = wave * 16 + l16;

  #pragma unroll
  for (int kb = 0; kb < 32; ++kb) {
    const int kbase = kb * 4 + khalf;
    f32x2 a = *(const f32x2*)&sA[aRow * 132 + kbase];
    #pragma unroll
    for (int t = 0; t < NT; ++t) {
      f32x2 b = *(const f32x2*)&sWt[(t * 16 + l16) * 132 + kbase];
      acc[t] = __builtin_amdgcn_wmma_f32_16x16x4_f32(
          /*neg_a=*/false, a, /*neg_b=*/false, b,
          /*c_mod=*/(short)0, acc[t], /*reuse_a=*/false, /*reuse_b=*/false);
    }
  }

  // C/D layout: VGPR r -> M = r (lanes 0-15) or r+8 (lanes 16-31); N = lane&15
  const int rowBase = rowBlk + wave * 16 + ((lane >> 4) << 3);
  #pragma unroll
  for (int t = 0; t < NT; ++t) {
    const int col = t * 16 + l16;
    if (col >= ncols) continue;
    const float bcol = bias[col];
    #pragma unroll
    for (int r = 0; r < 8; ++r) {
      const int row = rowBase + r;
      if (row < M) out[(size_t)row * ncols + col] = acc[t][r] + bcol;
    }
  }
}

// e[ei, :] = edge_weight[ei] * emb_e_W[0, :] + emb_e_b   (wave per edge)
__global__ __launch_bounds__(256)
void embed_e_kernel(const float* __restrict__ ew, const float* __restrict__ W,
                    const float* __restrict__ b, float* __restrict__ e) {
  const int gid = blockIdx.x * 256 + threadIdx.x;
  const int ei  = gid >> 5;
  const int f   = (gid & 31) << 2;
  if (ei >= NEDGE) return;
  const float w = ew[ei];
  float4 wv = *(const float4*)&W[f];
  float4 bv = *(const float4*)&b[f];
  float4 o;
  o.x = w * wv.x + bv.x; o.y = w * wv.y + bv.y;
  o.z = w * wv.z + bv.z; o.w = w * wv.w + bv.w;
  *(float4*)&e[(size_t)ei * HDIM + f] = o;
}

__global__ __launch_bounds__(256)
void zero_kernel(float* __restrict__ p, long long n4) {
  long long i = (long long)blockIdx.x * 256 + threadIdx.x;
  if (i < n4) *(float4*)&p[i * 4] = make_float4(0.f, 0.f, 0.f, 0.f);
}

// Fused edge phase: e_new = Dh[dst]+Eh[src]+Ce; sigma=sigmoid(e_new);
// scatter num += sigma*Bh[src], den += sigma; e = e_in + relu(e_new) in place.
// One wave per edge; lane covers 4 features (512B coalesced rows).
__global__ __launch_bounds__(256)
void edge_fused(const float* __restrict__ Ce, const float* __restrict__ Dh,
                const float* __restrict__ Eh, const float* __restrict__ Bh,
                const long long* __restrict__ eidx, float* __restrict__ e,
                float* __restrict__ num, float* __restrict__ den) {
  const int gid = blockIdx.x * 256 + threadIdx.x;
  const int ei  = gid >> 5;
  if (ei >= NEDGE) return;
  const int f = (gid & 31) << 2;
  const long long s = eidx[ei];
  const long long d = eidx[NEDGE + ei];
  const size_t eoff = (size_t)ei * HDIM + f;
  const size_t soff = (size_t)s * HDIM + f;
  const size_t doff = (size_t)d * HDIM + f;

  float4 ce  = *(const float4*)&Ce[eoff];
  float4 dh  = *(const float4*)&Dh[doff];
  float4 eh  = *(const float4*)&Eh[soff];
  float4 bh  = *(const float4*)&Bh[soff];
  float4 ein = *(const float4*)&e[eoff];

  float en[4]  = {dh.x + eh.x + ce.x, dh.y + eh.y + ce.y,
                  dh.z + eh.z + ce.z, dh.w + eh.w + ce.w};
  float bhv[4] = {bh.x, bh.y, bh.z, bh.w};
  float eiv[4] = {ein.x, ein.y, ein.z, ein.w};
  float eo[4];
  #pragma unroll
  for (int j = 0; j < 4; ++j) {
    float sig = 1.0f / (1.0f + __expf(-en[j]));
    atomicAdd(&num[doff + j], sig * bhv[j]);
    atomicAdd(&den[doff + j], sig);
    eo[j] = eiv[j] + fmaxf(en[j], 0.0f);
  }
  float4 o; o.x = eo[0]; o.y = eo[1]; o.z = eo[2]; o.w = eo[3];
  *(float4*)&e[eoff] = o;
}

// h = h_in + relu(Ah + num/(den+1e-6))   (in place)
__global__ __launch_bounds__(256)
void node_update(float* __restrict__ h, const float* __restrict__ Ah,
                 const float* __restrict__ num, const float* __restrict__ den) {
  const size_t idx = ((size_t)blockIdx.x * 256 + threadIdx.x) * 4;
  if (idx >= (size_t)NNODE * HDIM) return;
  float4 a  = *(const float4*)&Ah[idx];
  float4 nm = *(const float4*)&num[idx];
  float4 dn = *(const float4*)&den[idx];
  float4 hv = *(float4*)&h[idx];
  hv.x += fmaxf(a.x + nm.x / (dn.x + 1e-6f), 0.0f);
  hv.y += fmaxf(a.y + nm.y / (dn.y + 1e-6f), 0.0f);
  hv.z += fmaxf(a.z + nm.z / (dn.z + 1e-6f), 0.0f);
  hv.w += fmaxf(a.w + nm.w / (dn.w + 1e-6f), 0.0f);
  *(float4*)&h[idx] = hv;
}

// log_softmax over 40 classes; one wave per node, wave32 shuffle reduction.
__global__ __launch_bounds__(256)
void logsoftmax40(float* __restrict__ out) {
  const int gid  = blockIdx.x * 256 + threadIdx.x;
  const int node = gid >> 5;
  const int lane = gid & 31;
  if (node >= NNODE) return;
  float* row = out + (size_t)node * NCLASS;
  float x0 = row[lane];
  float x1 = (lane < NCLASS - 32) ? row[32 + lane] : -INFINITY;
  float m = fmaxf(x0, x1);
  #pragma unroll
  for (int o = 16; o > 0; o >>= 1) m = fmaxf(m, __shfl_xor(m, o, 32));
  float s = __expf(x0 - m) + ((lane < NCLASS - 32) ? __expf(x1 - m) : 0.0f);
  #pragma unroll
  for (int o = 16; o > 0; o >>= 1) s += __shfl_xor(s, o, 32);
  float lse = m + __logf(s);
  row[lane] = x0 - lse;
  if (lane < NCLASS - 32) row[32 + lane] = x1 - lse;
}

extern "C" void kernel_launch(void* const* d_in, const int* in_sizes, int n_in,
                              void* d_out, int out_size, void* d_ws, size_t ws_size,
                              hipStream_t stream) {
  (void)in_sizes; (void)n_in; (void)out_size; (void)ws_size;
  const float*      h_in    = (const float*)d_in[0];
  const long long*  eidx    = (const long long*)d_in[1];
  const float*      ew      = (const float*)d_in[2];
  const float*      emb_h_W = (const float*)d_in[3];
  const float*      emb_h_b = (const float*)d_in[4];
  const float*      emb_e_W = (const float*)d_in[5];
  const float*      emb_e_b = (const float*)d_in[6];
  const float*      WA = (const float*)d_in[7];
  const float*      bA = (const float*)d_in[8];
  const float*      WB = (const float*)d_in[9];
  const float*      bB = (const float*)d_in[10];
  const float*      WC = (const float*)d_in[11];
  const float*      bC = (const float*)d_in[12];
  const float*      WD = (const float*)d_in[13];
  const float*      bD = (const float*)d_in[14];
  const float*      WE = (const float*)d_in[15];
  const float*      bE = (const float*)d_in[16];
  const float*      mlpW = (const float*)d_in[17];
  const float*      mlpb = (const float*)d_in[18];

  const size_t NH = (size_t)NNODE * HDIM;
  const size_t EH = (size_t)NEDGE * HDIM;
  float* ws  = (float*)d_ws;
  float* h   = ws;           // NH
  float* Ah  = h   + NH;     // NH
  float* Bh  = Ah  + NH;     // NH
  float* Dh  = Bh  + NH;     // NH
  float* Eh  = Dh  + NH;     // NH
  float* num = Eh  + NH;     // NH
  float* den = num + NH;     // NH (adjacent to num: zeroed together)
  float* e   = den + NH;     // EH
  float* Ce  = e   + EH;     // EH

  const dim3 blk(256);
  const int gN = (NNODE + 127) / 128;   // 391
  const int gE = NEDGE / 128;           // 5000
  const int gEdgeWave = NEDGE / 8;      // 80000 (wave per edge)
  const int gNodeF4 = (int)((NH / 4 + 255) / 256);
  const long long zf4 = (long long)(2 * NH) / 4;
  const int gZero = (int)((zf4 + 255) / 256);

  // Input embeddings
  gemm_k128_wmma<8><<<dim3(gN), blk, 0, stream>>>(h_in, emb_h_W, emb_h_b, h, NNODE, HDIM);
  embed_e_kernel<<<dim3(gEdgeWave), blk, 0, stream>>>(ew, emb_e_W, emb_e_b, e);

  for (int i = 0; i < NLAYER; ++i) {
    const size_t wo = (size_t)i * HDIM * HDIM;
    const size_t bo = (size_t)i * HDIM;
    gemm_k128_wmma<8><<<dim3(gN), blk, 0, stream>>>(h, WA + wo, bA + bo, Ah, NNODE, HDIM);
    gemm_k128_wmma<8><<<dim3(gN), blk, 0, stream>>>(h, WB + wo, bB + bo, Bh, NNODE, HDIM);
    gemm_k128_wmma<8><<<dim3(gN), blk, 0, stream>>>(h, WD + wo, bD + bo, Dh, NNODE, HDIM);
    gemm_k128_wmma<8><<<dim3(gN), blk, 0, stream>>>(h, WE + wo, bE + bo, Eh, NNODE, HDIM);
    gemm_k128_wmma<8><<<dim3(gE), blk, 0, stream>>>(e, WC + wo, bC + bo, Ce, NEDGE, HDIM);
    zero_kernel<<<dim3(gZero), blk, 0, stream>>>(num, zf4);   // num+den contiguous
    edge_fused<<<dim3(gEdgeWave), blk, 0, stream>>>(Ce, Dh, Eh, Bh, eidx, e, num, den);
    node_update<<<dim3(gNodeF4), blk, 0, stream>>>(h, Ah, num, den);
  }

  // MLP + log_softmax
  gemm_k128_wmma<3><<<dim3(gN), blk, 0, stream>>>(h, mlpW, mlpb, (float*)d_out, NNODE, NCLASS);
  logsoftmax40<<<dim3((NNODE * 32 + 255) / 256), blk, 0, stream>>>((float*)d_out);
}